// MoTDiffFuncMod_27333171872221
// MI455X (gfx1250) — compile-verified
//
#include <hip/hip_runtime.h>

typedef __attribute__((ext_vector_type(2))) float v2f;
typedef __attribute__((ext_vector_type(8))) float v8f;

#define B_    4
#define S_    4096
#define D_    1024
#define OUT_  4096
#define NTOK  (B_ * S_)   // 16384 tokens

#define TILE_M   128
#define TILE_N   128
#define TILE_K   32
#define A_STRIDE 36       // 32 + pad: conflict-free column reads
#define B_STRIDE 136      // 128 + pad: half-wave K rows hit disjoint bank sets

__global__ __launch_bounds__(256)
void mot_expert_gemm(const float* __restrict__ X,
                     const int*   __restrict__ type_ids,
                     const float* __restrict__ W0, const float* __restrict__ b0,
                     const float* __restrict__ W1, const float* __restrict__ b1,
                     float* __restrict__ out)
{
    __shared__ float As[TILE_M * A_STRIDE];
    __shared__ float Bs[TILE_K * B_STRIDE];
    __shared__ int   Tid[TILE_M];

    const int e = blockIdx.z;
    const float* __restrict__ W    = e ? W1 : W0;
    const float* __restrict__ bias = e ? b1 : b0;
    float* __restrict__ outE = out + (size_t)e * (size_t)NTOK * OUT_;

    const int tokBase = blockIdx.y * TILE_M;
    const int colBase = blockIdx.x * TILE_N;

    const int tid  = threadIdx.x;
    const int lane = tid & 31;
    const int wave = tid >> 5;

    if (tid < TILE_M) Tid[tid] = type_ids[tokBase + tid];

    v8f acc[8];
#pragma unroll
    for (int j = 0; j < 8; ++j) acc[j] = (v8f){0.f,0.f,0.f,0.f,0.f,0.f,0.f,0.f};

    const int m16   = lane & 15;
    const int kpair = (lane >> 4) << 1;   // lanes 0-15 -> K{0,1}; lanes 16-31 -> K{2,3}

    for (int kc = 0; kc < D_; kc += TILE_K) {
        __syncthreads();
        // Cooperative load A tile: 128 rows x 32 floats (1024 float4, 4/thread)
#pragma unroll
        for (int it = 0; it < 4; ++it) {
            int id = tid + 256 * it;
            int r  = id >> 3;
            int c4 = id & 7;
            float4 v = *(const float4*)(X + (size_t)(tokBase + r) * D_ + kc + c4 * 4);
            *(float4*)(&As[r * A_STRIDE + c4 * 4]) = v;
        }
        // Cooperative load B tile: 32 rows x 128 floats (1024 float4, 4/thread)
#pragma unroll
        for (int it = 0; it < 4; ++it) {
            int id = tid + 256 * it;
            int r  = id >> 5;
            int c4 = id & 31;
            float4 v = *(const float4*)(W + (size_t)(kc + r) * OUT_ + colBase + c4 * 4);
            *(float4*)(&Bs[r * B_STRIDE + c4 * 4]) = v;
        }
        __syncthreads();

        const int rbase = wave * 16;
#pragma unroll
        for (int ks = 0; ks < TILE_K / 4; ++ks) {
            const int k0 = ks * 4;
            // A fragment (16x4 f32): lane's row = m16, K pair per half-wave
            v2f a;
            a.x = As[(rbase + m16) * A_STRIDE + k0 + kpair];
            a.y = As[(rbase + m16) * A_STRIDE + k0 + kpair + 1];
#pragma unroll
            for (int j = 0; j < 8; ++j) {
                // B fragment (4x16 f32): N striped across lanes, K pair per half-wave
                v2f bf;
                bf.x = Bs[(k0 + kpair)     * B_STRIDE + j * 16 + m16];
                bf.y = Bs[(k0 + kpair + 1) * B_STRIDE + j * 16 + m16];
                acc[j] = __builtin_amdgcn_wmma_f32_16x16x4_f32(
                    /*neg_a=*/false, a, /*neg_b=*/false, bf,
                    /*c_mod=*/(short)0, acc[j],
                    /*reuse_a=*/false, /*reuse_b=*/false);
            }
        }
    }

    // Epilogue: bias + modality mask. Output is 512 MB streamed once and never
    // re-read -> non-temporal stores so X/W (96 MB) stay resident in the 192 MB L2.
    const int rowHalf = (lane >> 4) * 8;  // lanes 16-31 hold M = r + 8
#pragma unroll
    for (int j = 0; j < 8; ++j) {
        const int n  = colBase + j * 16 + m16;
        const float bv = bias[n];
#pragma unroll
        for (int r = 0; r < 8; ++r) {
            const int rowInTile = wave * 16 + r + rowHalf;
            const float maskv = (Tid[rowInTile] == e) ? 1.0f : 0.0f;
            float* dst = outE + (size_t)(tokBase + rowInTile) * OUT_ + n;
            __builtin_nontemporal_store((acc[j][r] + bv) * maskv, dst);
        }
    }
}

extern "C" void kernel_launch(void* const* d_in, const int* in_sizes, int n_in,
                              void* d_out, int out_size, void* d_ws, size_t ws_size,
                              hipStream_t stream) {
    const float* X   = (const float*)d_in[0];   // hidden_states [B,S,D]
    const int*   tid = (const int*)  d_in[1];   // type_ids [B,S]
    const float* W0  = (const float*)d_in[2];   // [D,OUT]
    const float* b0  = (const float*)d_in[3];   // [OUT]
    const float* W1  = (const float*)d_in[4];   // [D,OUT]
    const float* b1  = (const float*)d_in[5];   // [OUT]
    float* out = (float*)d_out;                 // [2, B, S, OUT] concatenated

    dim3 grid(OUT_ / TILE_N, NTOK / TILE_M, 2);
    dim3 block(256);
    mot_expert_gemm<<<grid, block, 0, stream>>>(X, tid, W0, b0, W1, b1, out);
}